// K2GNNLayer_40432822125207
// MI455X (gfx1250) — compile-verified
//
#include <hip/hip_runtime.h>

typedef __attribute__((ext_vector_type(2))) float v2f;
typedef __attribute__((ext_vector_type(8))) float v8f;

#define D 128        // D_IN == D_OUT == 128
#define WT_STRIDE 132  // 132 mod 64 == 4 -> conflict-free b64 LDS reads

// ---------------------------------------------------------------------------
// Kernel 1: dual GEMM.  blockIdx.y == 0 -> XW + b (into d_out staging),
//           blockIdx.y == 1 -> XW_prop (into workspace).
// W is staged once per block into LDS *transposed* (Wt[n][k]) so each B
// fragment is one contiguous ds_load_b64.  Each wave computes a 16-row x
// 128-col stripe with V_WMMA_F32_16X16X4_F32 (exact fp32 math).
// 256 threads = 8 waves -> 128 rows per block.
// ---------------------------------------------------------------------------
__global__ __launch_bounds__(256) void gemm_dual_kernel(
    const float* __restrict__ X, const float* __restrict__ W,
    const float* __restrict__ Wp, const float* __restrict__ bias,
    float* __restrict__ outW, float* __restrict__ outWp, int n_nodes)
{
  __shared__ __align__(16) float Wt[D * WT_STRIDE];  // 67,584 B of 320 KB WGP LDS

  const bool prop   = (blockIdx.y != 0);
  const float* Wm   = prop ? Wp   : W;
  float*       dest = prop ? outWp : outW;

  // ---- Stage W transposed into LDS (uniform across block, before any exit) --
  // 16384 floats, 256 threads, float4 global reads; Wt[n*132 + k] = W[k*128+n]
  for (int i = threadIdx.x; i < (D * D) / 4; i += 256) {
    const float4 w4 = ((const float4*)Wm)[i];
    const int f = i * 4;
    const int k = f >> 7;       // row of W    (K index)
    const int n = f & (D - 1);  // column of W (N index)
    Wt[(n + 0) * WT_STRIDE + k] = w4.x;
    Wt[(n + 1) * WT_STRIDE + k] = w4.y;
    Wt[(n + 2) * WT_STRIDE + k] = w4.z;
    Wt[(n + 3) * WT_STRIDE + k] = w4.w;
  }
  __syncthreads();

  const int wave    = threadIdx.x >> 5;
  const int lane    = threadIdx.x & 31;
  const int rowbase = blockIdx.x * 128 + wave * 16;
  // n_nodes (100000) is a multiple of 16 -> a wave is either fully in-range
  // or fully out-of-range; exit keeps EXEC all-ones for the WMMA path.
  if (rowbase >= n_nodes) return;

  const int m     = lane & 15;        // A row / B,C column within tile
  const int khalf = (lane >> 4) * 2;  // K sub-pair selected by lane half
  const float* xrow = X + (size_t)(rowbase + m) * D;

  v8f c[8] = {};  // 8 N-tiles of 16x16 f32 accumulators

  for (int k = 0; k < D; k += 4) {
    const int kk = k + khalf;
    // A fragment: lane m holds row m, elements K=kk, kk+1 (one global b64)
    v2f a = *(const v2f*)(xrow + kk);
    #pragma unroll
    for (int t = 0; t < 8; ++t) {
      const int n = t * 16 + m;
      // B fragment: column n, rows K=kk,kk+1 -> contiguous in Wt (ds_load_b64)
      v2f bf = *(const v2f*)&Wt[n * WT_STRIDE + kk];
      c[t] = __builtin_amdgcn_wmma_f32_16x16x4_f32(
          /*neg_a=*/false, a, /*neg_b=*/false, bf,
          /*c_mod=*/(short)0, c[t], /*reuse_a=*/false, /*reuse_b=*/false);
    }
  }

  // C/D layout: VGPR v -> row (v + 8*(lane/16)), column (lane%16)
  const int rowoff = (lane >> 4) * 8;
  #pragma unroll
  for (int t = 0; t < 8; ++t) {
    const int col  = t * 16 + m;
    const float bv = prop ? 0.0f : bias[col];
    #pragma unroll
    for (int v = 0; v < 8; ++v) {
      const int row = rowbase + rowoff + v;
      dest[(size_t)row * D + col] = c[t][v] + bv;
    }
  }
}

// ---------------------------------------------------------------------------
// Kernel 2: edge gather + scatter-add.  One wave per edge; lane l handles
// columns [4l, 4l+4).  XW_prop (51 MB) and the accumulator (51 MB) are both
// L2-resident on MI455X (192 MB L2), so the float4 gathers and the
// global_atomic_add_f32 scatter run at L2 speed, not HBM speed.
// ---------------------------------------------------------------------------
__global__ __launch_bounds__(256) void edge_scatter_kernel(
    const float* __restrict__ XWp, const int* __restrict__ ref_a,
    const int* __restrict__ backref, float* __restrict__ acc, int n_edges)
{
  const int gid  = blockIdx.x * blockDim.x + threadIdx.x;
  const int e    = gid >> 5;
  const int lane = gid & 31;
  if (e >= n_edges) return;

  const int src = ref_a[e];
  const int dst = backref[e];
  const float4 v = ((const float4*)(XWp + (size_t)src * D))[lane];
  float* p = acc + (size_t)dst * D + (size_t)lane * 4;
  atomicAdd(p + 0, v.x);
  atomicAdd(p + 1, v.y);
  atomicAdd(p + 2, v.z);
  atomicAdd(p + 3, v.w);
}

// ---------------------------------------------------------------------------
// Kernel 3: in-place ReLU over X_out (bias + conv already accumulated).
// ---------------------------------------------------------------------------
__global__ __launch_bounds__(256) void relu_kernel(float* __restrict__ out, int n4)
{
  const int i = blockIdx.x * blockDim.x + threadIdx.x;
  if (i >= n4) return;
  float4* p = (float4*)out;
  float4 v = p[i];
  v.x = fmaxf(v.x, 0.0f);
  v.y = fmaxf(v.y, 0.0f);
  v.z = fmaxf(v.z, 0.0f);
  v.w = fmaxf(v.w, 0.0f);
  p[i] = v;
}

extern "C" void kernel_launch(void* const* d_in, const int* in_sizes, int n_in,
                              void* d_out, int out_size, void* d_ws, size_t ws_size,
                              hipStream_t stream) {
  // Input order per setup_inputs(): X, ref_a, backref, e_map, v_count, W, W_prop, b
  const float* X       = (const float*)d_in[0];
  const int*   ref_a   = (const int*)  d_in[1];
  const int*   backref = (const int*)  d_in[2];
  const float* W       = (const float*)d_in[5];
  const float* Wp      = (const float*)d_in[6];
  const float* bias    = (const float*)d_in[7];

  const int n_nodes = in_sizes[0] / D;   // 100000
  const int n_edges = in_sizes[1];       // 1600000

  float* Xout = (float*)d_out;           // X_out staging/accumulator
  float* XWp  = (float*)d_ws;            // XW_prop: n_nodes * 128 * 4 = 51.2 MB

  // 1) Dual GEMM: y=0 writes XW+b into d_out, y=1 writes XW_prop into ws.
  dim3 ggrid((n_nodes + 127) / 128, 2);
  gemm_dual_kernel<<<ggrid, dim3(256), 0, stream>>>(X, W, Wp, bias, Xout, XWp, n_nodes);

  // 2) Gather XW_prop rows at edge sources, atomic-add into edge targets.
  const int edge_threads = n_edges * 32;  // one wave per edge
  edge_scatter_kernel<<<(edge_threads + 255) / 256, 256, 0, stream>>>(
      XWp, ref_a, backref, Xout, n_edges);

  // 3) ReLU in place.
  const int n4 = n_nodes * D / 4;
  relu_kernel<<<(n4 + 255) / 256, 256, 0, stream>>>(Xout, n4);

  // 4) Pass-through tuple members (all int32, 4 bytes — same width as float).
  char* tail = (char*)d_out + (size_t)n_nodes * D * sizeof(float);
  const size_t ebytes = (size_t)n_edges * sizeof(int);
  hipMemcpyAsync(tail,              d_in[1], ebytes,      hipMemcpyDeviceToDevice, stream);
  hipMemcpyAsync(tail + ebytes,     d_in[2], ebytes,      hipMemcpyDeviceToDevice, stream);
  hipMemcpyAsync(tail + 2 * ebytes, d_in[3], ebytes,      hipMemcpyDeviceToDevice, stream);
  hipMemcpyAsync(tail + 3 * ebytes, d_in[4], sizeof(int), hipMemcpyDeviceToDevice, stream);
}